// SimpleNetwork_46153718563206
// MI455X (gfx1250) — compile-verified
//
#include <hip/hip_runtime.h>

#define VOCAB   100000
#define EMB     128
#define T_LEN   200
#define NBATCH  4096

typedef __attribute__((ext_vector_type(2))) float v2f;
typedef __attribute__((ext_vector_type(8))) float v8f;

// ---------------------------------------------------------------------------
// Kernel 1: embedding gather + masked mean pool.
// One wave32 per sample: 32 lanes x float4 == one 512B embedding row per load.
// Trip count (len) is wave-uniform -> no divergence. Prefetch 16 rows ahead
// to keep many random gathers in flight (latency-bound phase).
// ---------------------------------------------------------------------------
__global__ void emb_pool_kernel(const int* __restrict__ x,
                                const int* __restrict__ lengths,
                                const float* __restrict__ emb,
                                float* __restrict__ pooled) {
    const int wave = threadIdx.x >> 5;
    const int lane = threadIdx.x & 31;
    const int n = blockIdx.x * (blockDim.x >> 5) + wave;
    if (n >= NBATCH) return;

    const int  len = lengths[n];
    const int* xr  = x + n * T_LEN;
    const float* base = emb + lane * 4;          // this lane's 16B slice of a row

    float4 acc = make_float4(0.f, 0.f, 0.f, 0.f);

    #pragma unroll 4
    for (int t = 0; t < len; ++t) {
        const int row = xr[t];                   // wave-uniform broadcast load
        if (t + 16 < len) {
            // hide gather latency: global_prefetch_b8 of a future row
            __builtin_prefetch(base + (size_t)xr[t + 16] * EMB, 0, 0);
        }
        const float4 v = *reinterpret_cast<const float4*>(base + (size_t)row * EMB);
        acc.x += v.x; acc.y += v.y; acc.z += v.z; acc.w += v.w;
    }

    const float inv = 1.0f / (float)len;
    acc.x *= inv; acc.y *= inv; acc.z *= inv; acc.w *= inv;
    *reinterpret_cast<float4*>(pooled + (size_t)n * EMB + lane * 4) = acc;
}

// ---------------------------------------------------------------------------
// Kernel 2: C[M,Fout] = relu?(A[M,K] @ W[Fout,K]^T + b) via V_WMMA_F32_16X16X4_F32.
// One wave computes a 16x32 output tile: two accumulators share one A fragment,
// giving two independent WMMA chains (pipelines the matrix unit) and halving
// A-fragment traffic.
//   A frag (16x4):  lane l holds row m0+(l&15), K = k0 + 2*(l>>4) + {0,1}
//   B frag (4x16):  lane l holds col n?+(l&15), VGPR r <-> K = k0 + 2*(l>>4) + r
//   C frag (16x16): lane l col n?+(l&15); VGPR r <-> row m0 + 8*(l>>4) + r
// Out-of-range columns (only layer 4, Fout=2): B is loaded from a clamped
// in-bounds weight row; since output column n depends only on B column n and
// both live in lane (l&15)==n, the garbage stays in columns we never store.
// => inner loop is branch-free, EXEC all-1s at every WMMA.
// ---------------------------------------------------------------------------
__global__ void gemm_wmma_f32(const float* __restrict__ A,
                              const float* __restrict__ W,     // [Fout, K] row-major
                              const float* __restrict__ bias,  // [Fout]
                              float* __restrict__ C,           // [M, Fout]
                              int M, int K, int Fout, int tiles_n, int do_relu) {
    const int wave = threadIdx.x >> 5;
    const int lane = threadIdx.x & 31;
    const int tiles_m = M >> 4;
    const int tile = blockIdx.x * (blockDim.x >> 5) + wave;
    if (tile >= tiles_m * tiles_n) return;

    const int tm = tile % tiles_m;         // consecutive waves: same tn -> B reuse
    const int tn = tile / tiles_m;
    const int m0 = tm << 4;
    const int n0 = tn << 5;                // 32-wide output tile

    const int half = lane >> 4;            // 0: lanes 0-15, 1: lanes 16-31
    const int lr   = lane & 15;
    const int arow  = m0 + lr;
    const int bcol0 = n0 + lr;
    const int bcol1 = n0 + 16 + lr;

    // clamp to an in-bounds weight row; garbage only pollutes unstored columns
    const int cc0 = bcol0 < Fout ? bcol0 : (Fout - 1);
    const int cc1 = bcol1 < Fout ? bcol1 : (Fout - 1);

    const float* __restrict__ arp = A + (size_t)arow * K;
    const float* __restrict__ w0  = W + (size_t)cc0 * K;
    const float* __restrict__ w1  = W + (size_t)cc1 * K;

    v8f c0 = {0.f, 0.f, 0.f, 0.f, 0.f, 0.f, 0.f, 0.f};
    v8f c1 = {0.f, 0.f, 0.f, 0.f, 0.f, 0.f, 0.f, 0.f};

    #pragma unroll 4
    for (int k0 = 0; k0 < K; k0 += 4) {
        const int ka = k0 + 2 * half;      // 8-byte aligned (k0%4==0, half in {0,1})
        const v2f a  = *reinterpret_cast<const v2f*>(arp + ka);
        const v2f b0 = *reinterpret_cast<const v2f*>(w0 + ka);
        const v2f b1 = *reinterpret_cast<const v2f*>(w1 + ka);
        c0 = __builtin_amdgcn_wmma_f32_16x16x4_f32(false, a, false, b0,
                                                   (short)0, c0, false, false);
        c1 = __builtin_amdgcn_wmma_f32_16x16x4_f32(false, a, false, b1,
                                                   (short)0, c1, false, false);
    }

    const int row_base = m0 + 8 * half;
    if (bcol0 < Fout) {
        const float bv = bias[bcol0];
        #pragma unroll
        for (int r = 0; r < 8; ++r) {
            float v = c0[r] + bv;
            if (do_relu) v = fmaxf(v, 0.0f);
            C[(size_t)(row_base + r) * Fout + bcol0] = v;
        }
    }
    if (bcol1 < Fout) {
        const float bv = bias[bcol1];
        #pragma unroll
        for (int r = 0; r < 8; ++r) {
            float v = c1[r] + bv;
            if (do_relu) v = fmaxf(v, 0.0f);
            C[(size_t)(row_base + r) * Fout + bcol1] = v;
        }
    }
}

// ---------------------------------------------------------------------------
// Host-side launcher.
// Inputs (setup_inputs order): x, lengths, emb, W1, b1, W2, b2, W3, b3, W4, b4
// Output: [4096, 2] float32
// Workspace layout: pooled(2MB) | h1(8MB) | h2(4MB) | h3(2MB)  -> 16MB total
// ---------------------------------------------------------------------------
extern "C" void kernel_launch(void* const* d_in, const int* in_sizes, int n_in,
                              void* d_out, int out_size, void* d_ws, size_t ws_size,
                              hipStream_t stream) {
    const int*   x       = (const int*)d_in[0];
    const int*   lengths = (const int*)d_in[1];
    const float* emb     = (const float*)d_in[2];
    const float* W1 = (const float*)d_in[3];  const float* b1 = (const float*)d_in[4];
    const float* W2 = (const float*)d_in[5];  const float* b2 = (const float*)d_in[6];
    const float* W3 = (const float*)d_in[7];  const float* b3 = (const float*)d_in[8];
    const float* W4 = (const float*)d_in[9];  const float* b4 = (const float*)d_in[10];
    float* out = (float*)d_out;

    char* ws = (char*)d_ws;
    float* pooled = (float*)(ws);                                  // 4096*128*4 = 2 MB
    float* h1     = (float*)(ws + (size_t)2  * 1024 * 1024);       // 4096*512*4 = 8 MB
    float* h2     = (float*)(ws + (size_t)10 * 1024 * 1024);       // 4096*256*4 = 4 MB
    float* h3     = (float*)(ws + (size_t)14 * 1024 * 1024);       // 4096*128*4 = 2 MB

    // 1) gather + mean pool: 8 waves/block, one wave per sample
    {
        const int waves_per_block = 8;
        dim3 grid(NBATCH / waves_per_block), block(waves_per_block * 32);
        emb_pool_kernel<<<grid, block, 0, stream>>>(x, lengths, emb, pooled);
    }

    // GEMM launcher helper: 4 waves (16x32 tiles) per 128-thread block
    auto launch_gemm = [&](const float* A, const float* W, const float* b, float* C,
                           int K, int Fout, int relu) {
        const int tiles_m = NBATCH / 16;
        const int tiles_n = (Fout + 31) / 32;      // 32-wide tiles
        const int tiles   = tiles_m * tiles_n;
        const int tiles_per_block = 4;
        dim3 grid((tiles + tiles_per_block - 1) / tiles_per_block);
        dim3 block(tiles_per_block * 32);
        gemm_wmma_f32<<<grid, block, 0, stream>>>(A, W, b, C, NBATCH, K, Fout, tiles_n, relu);
    };

    // 2) h1 = relu(pooled @ W1^T + b1)   [4096,128]->[4096,512]
    launch_gemm(pooled, W1, b1, h1, 128, 512, 1);
    // 3) h2 = relu(h1 @ W2^T + b2)       [4096,512]->[4096,256]
    launch_gemm(h1, W2, b2, h2, 512, 256, 1);
    // 4) h3 = relu(h2 @ W3^T + b3)       [4096,256]->[4096,128]
    launch_gemm(h2, W3, b3, h3, 256, 128, 1);
    // 5) out = h3 @ W4^T + b4            [4096,128]->[4096,2]
    launch_gemm(h3, W4, b4, out, 128, 2, 0);
}